// PyramidAtt_10909216932448
// MI455X (gfx1250) — compile-verified
//
#include <hip/hip_runtime.h>

// ---------------------------------------------------------------------------
// PyramidAttention fused CDNA5 kernels (gfx1250, wave32)
//   - V_WMMA_F32_16X16X4_F32 for the fp32 GEMMs (exact fp32 accumulate)
//   - TENSOR_LOAD_TO_LDS (TDM) double-buffered A-tile staging with hardware
//     LDS padding (row 64 DWORDs + 4 DWORDs pad == conflict-free stride 68)
// B=512, H=8 (64 positions), E=2048, D=1024, A=512
// Key optimization: avg-pool commutes with the linear projection, so we
// project once (64 rows/batch) and pool the A=512-wide projected features.
// ---------------------------------------------------------------------------

typedef __attribute__((ext_vector_type(2))) float v2f;
typedef __attribute__((ext_vector_type(8))) float v8f;
typedef __attribute__((ext_vector_type(4))) unsigned int u32x4;
typedef __attribute__((ext_vector_type(8))) int i32x8;
typedef __attribute__((ext_vector_type(4))) int i32x4;

#define WMMA_F32(a, b, c) \
  __builtin_amdgcn_wmma_f32_16x16x4_f32(false, (a), false, (b), (short)0, (c), false, false)

#if defined(__AMDGCN__) && __has_builtin(__builtin_amdgcn_tensor_load_to_lds) && \
    __has_builtin(__builtin_amdgcn_s_wait_tensorcnt)
#define PA_HAVE_TDM 1
#else
#define PA_HAVE_TDM 0
#endif

#if __has_include(<hip/amd_detail/amd_gfx1250_TDM.h>)
#define PA_TDM_6ARG 1
#else
#define PA_TDM_6ARG 0
#endif

#if PA_HAVE_TDM
// DMA a 64-row x 64-float tile (row stride 2048 floats) from global into LDS,
// inserting 4 pad DWORDs after every 64 DWORDs (LDS row stride = 68 floats).
static __device__ __forceinline__ void tdm_load_tile(unsigned ldsAddr, const float* gptr) {
  unsigned long long ga = (unsigned long long)(uintptr_t)gptr;
  u32x4 g0;
  g0[0] = 1u;                                   // count=1, user descriptor
  g0[1] = ldsAddr;                              // LDS byte address
  g0[2] = (unsigned)ga;                         // global_addr[31:0]
  g0[3] = (unsigned)(ga >> 32) | (2u << 30);    // global_addr[56:32] | type=2
  i32x8 g1;
  g1[0] = (int)((2u << 16)      // data_size = 4 bytes
              | (1u << 20)      // pad_enable
              | (5u << 22)      // pad_interval: 64 DWORDs
              | (3u << 25));    // pad_amount:   4 DWORDs
  g1[1] = (int)(2048u << 16);                   // tensor_dim0[15:0] = 2048
  g1[2] = (int)((2048u >> 16) | (64u << 16));   // dim0[31:16] | tensor_dim1 = 64
  g1[3] = (int)(64u << 16);                     // dim1[31:16]=0 | tile_dim0 = 64
  g1[4] = (int)64u;                             // tile_dim1 = 64, tile_dim2 = 0
  g1[5] = (int)2048;                            // tensor_dim0_stride[31:0]
  g1[6] = 0;                                    // stride0[47:32] | dim1_stride lo
  g1[7] = 0;
  i32x4 z4 = {0, 0, 0, 0};
#if PA_TDM_6ARG
  i32x8 z8 = {0, 0, 0, 0, 0, 0, 0, 0};
  __builtin_amdgcn_tensor_load_to_lds(g0, g1, z4, z4, z8, 0);
#else
  __builtin_amdgcn_tensor_load_to_lds(g0, g1, z4, z4, 0);
#endif
}
#endif  // PA_HAVE_TDM

static __device__ __forceinline__ float bilin(const float* v, int W, float sy, float sx) {
  // half-pixel bilinear with clamped taps (== torch align_corners=False)
  int iy = (int)floorf(sy); float fy = sy - (float)iy;
  int ix = (int)floorf(sx); float fx = sx - (float)ix;
  int y0 = min(max(iy, 0), W - 1), y1 = min(max(iy + 1, 0), W - 1);
  int x0 = min(max(ix, 0), W - 1), x1 = min(max(ix + 1, 0), W - 1);
  float v00 = v[y0 * W + x0], v01 = v[y0 * W + x1];
  float v10 = v[y1 * W + x0], v11 = v[y1 * W + x1];
  return (v00 * (1.f - fx) + v01 * fx) * (1.f - fy) +
         (v10 * (1.f - fx) + v11 * fx) * fy;
}

// ---------------------------------------------------------------------------
// Kernel 1: att2[b][a] = decoder[b][:] @ W_dec[:, a] + b_dec[a]
// C = [512 x 512], K = 1024.  One wave computes a 16x32 C-tile (2 WMMA tiles).
// ---------------------------------------------------------------------------
__global__ __launch_bounds__(256) void att2_gemm(
    const float* __restrict__ dec, const float* __restrict__ Wdec,
    const float* __restrict__ bdec, float* __restrict__ att2) {
  constexpr int D = 1024, A = 512;
  const int tid = threadIdx.x;
  const int wave = tid >> 5, lane = tid & 31;
  const int g = blockIdx.x * 8 + wave;   // 0..511
  const int tr = g >> 4;                 // C row-tile 0..31
  const int tcBase = (g & 15) * 32;      // C col base (2 tiles of 16)
  const int m = lane & 15, kh = lane >> 4;

  v8f c0 = {}, c1 = {};
  const float* __restrict__ arow = dec + (size_t)(tr * 16 + m) * D;
  const float* __restrict__ b0p = Wdec + tcBase + m;
  const float* __restrict__ b1p = Wdec + tcBase + 16 + m;
  for (int k = 0; k < D; k += 4) {
    const int kr = k + 2 * kh;
    v2f a;   a.x = arow[kr];          a.y = arow[kr + 1];
    v2f bb0; bb0.x = b0p[kr * A];     bb0.y = b0p[(kr + 1) * A];
    v2f bb1; bb1.x = b1p[kr * A];     bb1.y = b1p[(kr + 1) * A];
    c0 = WMMA_F32(a, bb0, c0);
    c1 = WMMA_F32(a, bb1, c1);
  }
  const int col = tcBase + m;
#pragma unroll
  for (int r = 0; r < 8; ++r) {
    const int row = tr * 16 + r + 8 * kh;
    att2[row * A + col]      = c0[r] + bdec[col];
    att2[row * A + col + 16] = c1[r] + bdec[col + 16];
  }
}

// ---------------------------------------------------------------------------
// Kernel 2: one workgroup (8 wave32) per batch element.
//   y = x[b] (64x2048) @ W_enc (2048x512)  via WMMA f32, A-chunked by 128,
//   A tiles staged by TDM (double buffered); pooled/tanh/dot logits per chunk;
//   then upsample + softmax + context vector.
// ---------------------------------------------------------------------------
__global__ __launch_bounds__(256) void pyratt_fused(
    const float* __restrict__ x, const float* __restrict__ Wenc,
    const float* __restrict__ benc, const float* __restrict__ att2,
    const float* __restrict__ wfull, const float* __restrict__ bfull,
    float* __restrict__ outCtx, float* __restrict__ outAlpha) {
  constexpr int E = 2048, A = 512, NPOS = 64;
  constexpr int ACH = 128, NCH = A / ACH;          // a-chunking
  constexpr int KBLK = 64, KPAD = 68, YPAD = 132;  // padded LDS strides

#if PA_HAVE_TDM
  __shared__ float aSh[2][NPOS * KPAD];  // double-buffered staged A tiles
#else
  __shared__ float aSh[1][NPOS * KPAD];
#endif
  __shared__ float ySh[NPOS * YPAD];     // projected chunk (~33.8 KB)
  __shared__ float cal1[64], cal2[49], cal3[25];
  __shared__ float bwSh[ACH], wfSh[ACH];
  __shared__ float attSh[64], alphaSh[64], red[2];

  const int b = blockIdx.x;
  const int tid = threadIdx.x;
  const int wave = tid >> 5, lane = tid & 31;
  const int m = lane & 15, kh = lane >> 4;
  const float* __restrict__ xb = x + (size_t)b * NPOS * E;
  const float* __restrict__ att2b = att2 + b * A;

#if PA_HAVE_TDM
  const unsigned aOff0 = (unsigned)(uintptr_t)&aSh[0][0];  // LDS byte offsets
  const unsigned aOff1 = (unsigned)(uintptr_t)&aSh[1][0];
#endif

  if (tid < 64) cal1[tid] = bfull[0];
  if (tid < 49) cal2[tid] = bfull[0];
  if (tid < 25) cal3[tid] = bfull[0];

  for (int ch = 0; ch < NCH; ++ch) {
    const int aBase = ch * ACH;
    const int acol = aBase + wave * 16 + m;   // this wave's B/N column
    v8f acc0 = {}, acc1 = {}, acc2v = {}, acc3v = {};

    __syncthreads();   // everyone done with previous chunk's LDS use
#if PA_HAVE_TDM
    if (wave == 0) tdm_load_tile(aOff0, xb);  // prologue: tile kb=0 -> buf0
#endif

    for (int kb = 0; kb < E; kb += KBLK) {
#if PA_HAVE_TDM
      const int buf = (kb >> 6) & 1;
      if (wave == 0) __builtin_amdgcn_s_wait_tensorcnt(0);  // buf ready
      __syncthreads();                                       // publish to WG
      if (wave == 0 && (kb + KBLK) < E)                      // prefetch next
        tdm_load_tile(buf ? aOff0 : aOff1, xb + kb + KBLK);
      const float* __restrict__ aBuf = &aSh[buf][0];
#else
      __syncthreads();
      // cooperative stage of x[b][:, kb:kb+64] -> aSh (coalesced float4)
#pragma unroll
      for (int j = 0; j < 4; ++j) {
        int f = j * 256 + tid;          // 0..1023 float4 slots
        int n = f >> 4;
        int k4 = (f & 15) << 2;
        const float4 v = *reinterpret_cast<const float4*>(xb + n * E + kb + k4);
        float* dst = &aSh[0][n * KPAD + k4];
        dst[0] = v.x; dst[1] = v.y; dst[2] = v.z; dst[3] = v.w;
      }
      __syncthreads();
      const float* __restrict__ aBuf = &aSh[0][0];
#endif

      const float* __restrict__ bp = Wenc + acol;
#pragma unroll 4
      for (int k0 = 0; k0 < KBLK; k0 += 4) {
        const int krow = kb + k0 + 2 * kh;
        v2f bf; bf.x = bp[krow * A]; bf.y = bp[(krow + 1) * A];
        const int ak = k0 + 2 * kh;
        v2f a0 = *reinterpret_cast<const v2f*>(&aBuf[(0 * 16 + m) * KPAD + ak]);
        v2f a1 = *reinterpret_cast<const v2f*>(&aBuf[(1 * 16 + m) * KPAD + ak]);
        v2f a2 = *reinterpret_cast<const v2f*>(&aBuf[(2 * 16 + m) * KPAD + ak]);
        v2f a3 = *reinterpret_cast<const v2f*>(&aBuf[(3 * 16 + m) * KPAD + ak]);
        acc0  = WMMA_F32(a0, bf, acc0);
        acc1  = WMMA_F32(a1, bf, acc1);
        acc2v = WMMA_F32(a2, bf, acc2v);
        acc3v = WMMA_F32(a3, bf, acc3v);
      }
    }

    // spill accumulators -> ySh  (C layout: vgpr r -> M=r (+8 for hi lanes))
    {
      const int col = wave * 16 + m;
#pragma unroll
      for (int r = 0; r < 8; ++r) {
        ySh[(0 * 16 + r + 8 * kh) * YPAD + col] = acc0[r];
        ySh[(1 * 16 + r + 8 * kh) * YPAD + col] = acc1[r];
        ySh[(2 * 16 + r + 8 * kh) * YPAD + col] = acc2v[r];
        ySh[(3 * 16 + r + 8 * kh) * YPAD + col] = acc3v[r];
      }
    }
    if (tid < ACH) {
      bwSh[tid] = benc[aBase + tid] + att2b[aBase + tid];
      wfSh[tid] = wfull[aBase + tid];
    }
    __syncthreads();

    // ---- partial logits for this a-chunk (pool AFTER projection) ----
    {  // level 1: 64 positions, 4 lanes each over 32 a's
      const int p = tid >> 2, q = tid & 3;
      const float* yr = &ySh[p * YPAD];
      float s = 0.f;
      for (int ac = q * 32; ac < q * 32 + 32; ++ac)
        s += tanhf(yr[ac] + bwSh[ac]) * wfSh[ac];
      s += __shfl_xor(s, 1, 32);
      s += __shfl_xor(s, 2, 32);
      if (q == 0) cal1[p] += s;
    }
    if (tid < 196) {  // level 2: 2x2 mean pooling -> 7x7
      const int p = tid >> 2, q = tid & 3;
      const int i = p / 7, j = p % 7;
      const float* y00 = &ySh[(i * 8 + j) * YPAD];
      float s = 0.f;
      for (int ac = q * 32; ac < q * 32 + 32; ++ac) {
        float v = 0.25f * (y00[ac] + y00[YPAD + ac] +
                           y00[8 * YPAD + ac] + y00[9 * YPAD + ac]);
        s += tanhf(v + bwSh[ac]) * wfSh[ac];
      }
      s += __shfl_xor(s, 1, 32);
      s += __shfl_xor(s, 2, 32);
      if (q == 0) cal2[p] += s;
    }
    if (tid < 100) {  // level 3: 4x4 mean pooling -> 5x5
      const int p = tid >> 2, q = tid & 3;
      const int i = p / 5, j = p % 5;
      const float* y00 = &ySh[(i * 8 + j) * YPAD];
      float s = 0.f;
      for (int ac = q * 32; ac < q * 32 + 32; ++ac) {
        float v = 0.f;
#pragma unroll
        for (int dy = 0; dy < 4; ++dy)
#pragma unroll
          for (int dx = 0; dx < 4; ++dx)
            v += y00[(dy * 8 + dx) * YPAD + ac];
        s += tanhf(v * (1.f / 16.f) + bwSh[ac]) * wfSh[ac];
      }
      s += __shfl_xor(s, 1, 32);
      s += __shfl_xor(s, 2, 32);
      if (q == 0) cal3[p] += s;
    }
    __syncthreads();
  }

  // ---- bilinear upsample (half-pixel, clamped) + combine ----
  if (tid < 64) {
    const int oy = tid >> 3, ox = tid & 7;
    float u2 = bilin(cal2, 7, (oy + 0.5f) * 0.875f - 0.5f, (ox + 0.5f) * 0.875f - 0.5f);
    float u3 = bilin(cal3, 5, (oy + 0.5f) * 0.625f - 0.5f, (ox + 0.5f) * 0.625f - 0.5f);
    attSh[tid] = cal1[tid] + u2 + u3;
  }
  __syncthreads();
  if (tid == 0) {
    float mx = attSh[0];
    for (int i = 1; i < 64; ++i) mx = fmaxf(mx, attSh[i]);
    red[0] = mx;
  }
  __syncthreads();
  if (tid < 64) alphaSh[tid] = expf(attSh[tid] - red[0]);
  __syncthreads();
  if (tid == 0) {
    float s = 0.f;
    for (int i = 0; i < 64; ++i) s += alphaSh[i];
    red[1] = 1.f / s;
  }
  __syncthreads();
  if (tid < 64) {
    float al = alphaSh[tid] * red[1];
    alphaSh[tid] = al;
    outAlpha[b * 64 + tid] = al;
  }
  __syncthreads();

  // ---- context vector: coalesced re-read of x[b] ----
  for (int e = tid; e < E; e += 256) {
    float s = 0.f;
#pragma unroll 8
    for (int n = 0; n < 64; ++n) s += alphaSh[n] * xb[n * E + e];
    outCtx[(size_t)b * E + e] = s;
  }
}

// ---------------------------------------------------------------------------
extern "C" void kernel_launch(void* const* d_in, const int* in_sizes, int n_in,
                              void* d_out, int out_size, void* d_ws, size_t ws_size,
                              hipStream_t stream) {
  (void)in_sizes; (void)n_in; (void)out_size; (void)ws_size;
  const float* encoder = (const float*)d_in[0];  // [512,8,8,2048]
  const float* decoder = (const float*)d_in[1];  // [512,1024]
  const float* Wenc    = (const float*)d_in[2];  // [2048,512]
  const float* benc    = (const float*)d_in[3];  // [512]
  const float* Wdec    = (const float*)d_in[4];  // [1024,512]
  const float* bdec    = (const float*)d_in[5];  // [512]
  const float* wfull   = (const float*)d_in[6];  // [512]
  const float* bfull   = (const float*)d_in[7];  // [1]

  float* att2     = (float*)d_ws;                // 512*512 f32 = 1 MB scratch
  float* outCtx   = (float*)d_out;               // [512,2048]
  float* outAlpha = outCtx + 512 * 2048;         // [512,64]

  att2_gemm<<<64, 256, 0, stream>>>(decoder, Wdec, bdec, att2);
  pyratt_fused<<<512, 256, 0, stream>>>(encoder, Wenc, benc, att2,
                                        wfull, bfull, outCtx, outAlpha);
}